// ResGATHet_30047591203151
// MI455X (gfx1250) — compile-verified
//
#include <hip/hip_runtime.h>
#include <math.h>

typedef __attribute__((ext_vector_type(2))) float v2f;
typedef __attribute__((ext_vector_type(8))) float v8f;

#define F_IN    256
#define F_OUT   128
#define NHEAD   4
#define DHEAD   32
#define TLAYERS 2

// ---------------------------------------------------------------------------
// Ordered-uint encoding so float max can use integer atomicMax.
// enc is monotonic: enc(a) < enc(b)  <=>  a < b. enc of any float > 0u,
// so initializing the buffer to 0u acts as "-infinity".
// ---------------------------------------------------------------------------
__device__ __forceinline__ unsigned enc_f32(float f) {
    unsigned u = __float_as_uint(f);
    return u ^ ((u & 0x80000000u) ? 0xFFFFFFFFu : 0x80000000u);
}
__device__ __forceinline__ float dec_f32(unsigned u) {
    return __uint_as_float(u ^ ((u & 0x80000000u) ? 0x80000000u : 0xFFFFFFFFu));
}

// ---------------------------------------------------------------------------
// One wave (32 lanes) computes a 32-row x 128-col output block as two 16-row
// WMMA tiles that SHARE every B fragment: each B load feeds two
// V_WMMA_F32_16X16X4_F32 ops (different A fragments / accumulators), halving
// the B traffic per FLOP vs one tile per wave.
//
// fp32 WMMA fragment layout (CDNA5 ISA 7.12.2):
//   A 16x4:  lane&15 = M row; (lane>>4)*2 + {0,1} = K index (2 VGPRs)
//   B 4x16:  lane&15 = N col; (lane>>4)*2 + {0,1} = K index (2 VGPRs)
//   C/D 16x16: VGPR r, lane L -> m = r + (L>=16 ? 8 : 0), n = L&15
// ---------------------------------------------------------------------------
template <int K>
__device__ __forceinline__ void gemm32_tile(
    const float* __restrict__ A0, const float* __restrict__ A1, int lda,
    const float* __restrict__ B, // K x 128 row-major
    const float* __restrict__ bias,
    float* __restrict__ C0, float* __restrict__ C1, int ldc) {
    const int lane = threadIdx.x & 31;
    const int nlo  = lane & 15;
    const int khi  = (lane >> 4) << 1; // 0 or 2

    v8f acc0[8] = {};
    v8f acc1[8] = {};
    for (int k = 0; k < K; k += 4) {
        const float* ap0 = A0 + (size_t)nlo * lda + (k + khi);
        const float* ap1 = A1 + (size_t)nlo * lda + (k + khi);
        v2f a0; a0.x = ap0[0]; a0.y = ap0[1];
        v2f a1; a1.x = ap1[0]; a1.y = ap1[1];
#pragma unroll
        for (int nt = 0; nt < 8; ++nt) {
            const float* bp = B + (size_t)(k + khi) * F_OUT + nt * 16 + nlo;
            v2f b;
            b.x = bp[0];
            b.y = bp[F_OUT];
            acc0[nt] = __builtin_amdgcn_wmma_f32_16x16x4_f32(
                false, a0, false, b, (short)0, acc0[nt], false, false);
            acc1[nt] = __builtin_amdgcn_wmma_f32_16x16x4_f32(
                false, a1, false, b, (short)0, acc1[nt], false, false);
        }
    }

    const int mrow0 = (lane >> 4) * 8;
#pragma unroll
    for (int nt = 0; nt < 8; ++nt) {
        const int   n  = nt * 16 + nlo;
        const float bv = bias[n];
#pragma unroll
        for (int r = 0; r < 8; ++r) {
            C0[(size_t)(mrow0 + r) * ldc + n] = acc0[nt][r] + bv;
            C1[(size_t)(mrow0 + r) * ldc + n] = acc1[nt][r] + bv;
        }
    }
}

// ---- embedding: x = concat(x_user @ Wu + bu, x_item @ Wi + bi) -------------
// Each 16-row subtile lies entirely on one side (50000 is a multiple of 16),
// so A source / weights are picked per subtile.
__global__ __launch_bounds__(32) void k_embed(const float* __restrict__ xu,
                                              const float* __restrict__ xi,
                                              const float* __restrict__ wu,
                                              const float* __restrict__ bu,
                                              const float* __restrict__ wi,
                                              const float* __restrict__ bi,
                                              float* __restrict__ x, int n_user) {
    const int r0 = blockIdx.x * 32;
    const int r1 = r0 + 16;
    const float* A0 = (r0 < n_user) ? xu + (size_t)r0 * F_IN
                                    : xi + (size_t)(r0 - n_user) * F_IN;
    const float* A1 = (r1 < n_user) ? xu + (size_t)r1 * F_IN
                                    : xi + (size_t)(r1 - n_user) * F_IN;
    // Weights differ per side only when the 32-row block straddles the
    // boundary; since both subtiles use the same B in gemm32_tile, split the
    // straddling block into two 16-row passes with a degenerate second tile.
    const bool side0 = r0 < n_user, side1 = r1 < n_user;
    if (side0 == side1) {
        gemm32_tile<F_IN>(A0, A1, F_IN, side0 ? wu : wi, side0 ? bu : bi,
                          x + (size_t)r0 * F_OUT, x + (size_t)r1 * F_OUT, F_OUT);
    } else {
        // straddle: run each subtile against its own weights (A repeated,
        // second output aliased to the same rows -> harmless duplicate work)
        gemm32_tile<F_IN>(A0, A0, F_IN, wu, bu,
                          x + (size_t)r0 * F_OUT, x + (size_t)r0 * F_OUT, F_OUT);
        gemm32_tile<F_IN>(A1, A1, F_IN, wi, bi,
                          x + (size_t)r1 * F_OUT, x + (size_t)r1 * F_OUT, F_OUT);
    }
}

// ---- generic 128-K GEMM (h = x@lin_w + b, residual = x@res_w + b) ----------
__global__ __launch_bounds__(32) void k_gemm128(const float* __restrict__ A,
                                                const float* __restrict__ W,
                                                const float* __restrict__ bias,
                                                float* __restrict__ C, int ldc) {
    const int r0 = blockIdx.x * 32;
    gemm32_tile<F_OUT>(A + (size_t)r0 * F_OUT, A + (size_t)(r0 + 16) * F_OUT, F_OUT,
                       W, bias,
                       C + (size_t)r0 * ldc, C + (size_t)(r0 + 16) * ldc, ldc);
}

// ---- fold self_w/self_b through a_self: cvec[d] = sum_e W[d][e]*a_self[e] --
__global__ void k_cvec(const float* __restrict__ self_w,
                       const float* __restrict__ self_b,
                       const float* __restrict__ att_w,
                       float* __restrict__ cvec, float* __restrict__ c0) {
    const int t = blockIdx.x;
    const int d = threadIdx.x; // 32 threads
    const float* aself = att_w + t * 3 * DHEAD + 2 * DHEAD;
    float c = 0.f;
    for (int e = 0; e < DHEAD; ++e)
        c += self_w[t * DHEAD * DHEAD + d * DHEAD + e] * aself[e];
    cvec[t * DHEAD + d] = c;
    if (d == 0) {
        float s = 0.f;
        for (int e = 0; e < DHEAD; ++e) s += self_b[t * DHEAD + e] * aself[e];
        c0[t] = s;
    }
}

// ---- per-(node,head) attention scores --------------------------------------
__global__ void k_scores(const float* __restrict__ h, const float* __restrict__ att_w,
                         const float* __restrict__ cvec, const float* __restrict__ c0,
                         float* __restrict__ ssrc, float* __restrict__ sdst,
                         int n_total, int t) {
    const int tid = blockIdx.x * blockDim.x + threadIdx.x;
    if (tid >= n_total * NHEAD) return;
    const int n = tid >> 2, hh = tid & 3;
    const float* hp   = h + (size_t)n * F_OUT + hh * DHEAD;
    const float* asrc = att_w + t * 3 * DHEAD;
    const float* adst = asrc + DHEAD;
    const float* cv   = cvec + t * DHEAD;
    float s1 = 0.f, s2 = 0.f;
#pragma unroll
    for (int d = 0; d < DHEAD; ++d) {
        const float hv = hp[d];
        s1 += hv * (asrc[d] + cv[d]);
        s2 += hv * adst[d];
    }
    ssrc[tid] = s1 + c0[t];
    sdst[tid] = s2;
}

// ---- init segment-max / segment-sum buffers --------------------------------
__global__ void k_init(unsigned* __restrict__ menc, float* __restrict__ denom, int count) {
    const int i = blockIdx.x * blockDim.x + threadIdx.x;
    if (i < count) { menc[i] = 0u; denom[i] = 0.f; }
}

// ---- pass A: logits + segment max over src ---------------------------------
__global__ void k_edge_max(const int* __restrict__ src, const int* __restrict__ dst,
                           const float* __restrict__ ssrc, const float* __restrict__ sdst,
                           const float* __restrict__ att_b, int t,
                           float* __restrict__ logits, unsigned* __restrict__ menc, int e4) {
    const int i = blockIdx.x * blockDim.x + threadIdx.x;
    if (i >= e4) return;
    const int e = i >> 2, hh = i & 3;
    const int s = src[e], d = dst[e];
    float l = ssrc[s * NHEAD + hh] + sdst[d * NHEAD + hh] + att_b[t];
    l = (l > 0.f) ? l : 0.2f * l; // leaky_relu(0.2)
    logits[i] = l;
    atomicMax(menc + s * NHEAD + hh, enc_f32(l));
}

// ---- pass B: exp(logit - max) + segment sum over src -----------------------
__global__ void k_edge_exp(const int* __restrict__ src,
                           float* __restrict__ logits,
                           const unsigned* __restrict__ menc,
                           float* __restrict__ denom, int e4) {
    const int i = blockIdx.x * blockDim.x + threadIdx.x;
    if (i >= e4) return;
    const int e = i >> 2, hh = i & 3;
    const int s = src[e];
    const float m  = dec_f32(menc[s * NHEAD + hh]);
    const float ex = expf(logits[i] - m);
    logits[i] = ex; // reuse buffer: now holds e-values
    atomicAdd(denom + s * NHEAD + hh, ex);
}

// ---- pass C: scatter weighted messages into output -------------------------
// One wave per edge; lane handles 4 consecutive channels (all in one head).
__global__ __launch_bounds__(256) void k_edge_scatter(
    const int* __restrict__ src, const int* __restrict__ dst,
    const float* __restrict__ evals, const float* __restrict__ denom,
    const float* __restrict__ ew, const float* __restrict__ h,
    float* __restrict__ out, int E, int t) {
    const int w    = threadIdx.x >> 5;
    const int lane = threadIdx.x & 31;
    const int e    = blockIdx.x * 8 + w;
    if (e >= E) return;
    const int s  = src[e], d = dst[e];
    const int hh = lane >> 3; // 8 lanes (32 channels) per head
    const float alpha = evals[e * NHEAD + hh] / denom[s * NHEAD + hh];
    const float wgt   = ew[e] * alpha;
    const float4 hv = ((const float4*)(h + (size_t)s * F_OUT))[lane];
    float* op = out + (size_t)d * (TLAYERS * F_OUT) + t * F_OUT + lane * 4;
    atomicAdd(op + 0, wgt * hv.x);
    atomicAdd(op + 1, wgt * hv.y);
    atomicAdd(op + 2, wgt * hv.z);
    atomicAdd(op + 3, wgt * hv.w);
}

// ---------------------------------------------------------------------------
extern "C" void kernel_launch(void* const* d_in, const int* in_sizes, int n_in,
                              void* d_out, int out_size, void* d_ws, size_t ws_size,
                              hipStream_t stream) {
    const float* x_user     = (const float*)d_in[0];
    const float* x_item     = (const float*)d_in[1];
    const int*   edge_index = (const int*)d_in[2];
    const float* edge_wt    = (const float*)d_in[3];
    const float* emb_w_user = (const float*)d_in[4];
    const float* emb_b_user = (const float*)d_in[5];
    const float* emb_w_item = (const float*)d_in[6];
    const float* emb_b_item = (const float*)d_in[7];
    const float* lin_w      = (const float*)d_in[8];
    const float* lin_b      = (const float*)d_in[9];
    const float* self_w     = (const float*)d_in[10];
    const float* self_b     = (const float*)d_in[11];
    const float* att_w      = (const float*)d_in[12];
    const float* att_b      = (const float*)d_in[13];
    const float* res_w      = (const float*)d_in[14];
    const float* res_b      = (const float*)d_in[15];
    float* out = (float*)d_out;

    const int n_user  = in_sizes[0] / F_IN;
    const int n_item  = in_sizes[1] / F_IN;
    const int n_total = n_user + n_item;
    const int E       = in_sizes[3] / TLAYERS;

    // ---- carve workspace ----
    char*  ws  = (char*)d_ws;
    size_t off = 0;
    auto carve = [&](size_t bytes) {
        char* p = ws + off;
        off = (off + bytes + 255) & ~(size_t)255;
        return p;
    };
    float*    x      = (float*)carve((size_t)n_total * F_OUT * sizeof(float));
    float*    h      = (float*)carve((size_t)n_total * F_OUT * sizeof(float));
    float*    ssrc   = (float*)carve((size_t)n_total * NHEAD * sizeof(float));
    float*    sdst   = (float*)carve((size_t)n_total * NHEAD * sizeof(float));
    float*    logits = (float*)carve((size_t)E * NHEAD * sizeof(float));
    unsigned* menc   = (unsigned*)carve((size_t)n_total * NHEAD * sizeof(unsigned));
    float*    denom  = (float*)carve((size_t)n_total * NHEAD * sizeof(float));
    float*    cvec   = (float*)carve(TLAYERS * DHEAD * sizeof(float));
    float*    c0     = (float*)carve(TLAYERS * sizeof(float));

    const int mtiles32 = (n_total + 31) / 32; // 100000/32 = 3125 exact

    // embedding GEMM (WMMA fp32)
    k_embed<<<mtiles32, 32, 0, stream>>>(x_user, x_item, emb_w_user, emb_b_user,
                                         emb_w_item, emb_b_item, x, n_user);
    // fold self-transform into attention vectors
    k_cvec<<<TLAYERS, DHEAD, 0, stream>>>(self_w, self_b, att_w, cvec, c0);

    for (int t = 0; t < TLAYERS; ++t) {
        const int* src = edge_index + (size_t)t * 2 * E;
        const int* dst = src + E;

        // residual = x @ res_w + res_b, written straight into out[:, t, :]
        k_gemm128<<<mtiles32, 32, 0, stream>>>(x, res_w + (size_t)t * F_OUT * F_OUT,
                                               res_b + t * F_OUT, out + t * F_OUT,
                                               TLAYERS * F_OUT);
        // h = x @ lin_w + lin_b
        k_gemm128<<<mtiles32, 32, 0, stream>>>(x, lin_w + (size_t)t * F_OUT * F_OUT,
                                               lin_b + t * F_OUT, h, F_OUT);

        const int nh = n_total * NHEAD;
        k_scores<<<(nh + 255) / 256, 256, 0, stream>>>(h, att_w, cvec, c0, ssrc, sdst,
                                                       n_total, t);
        k_init<<<(nh + 255) / 256, 256, 0, stream>>>(menc, denom, nh);

        const int e4 = E * NHEAD;
        k_edge_max<<<(e4 + 255) / 256, 256, 0, stream>>>(src, dst, ssrc, sdst, att_b, t,
                                                         logits, menc, e4);
        k_edge_exp<<<(e4 + 255) / 256, 256, 0, stream>>>(src, logits, menc, denom, e4);
        k_edge_scatter<<<(E + 7) / 8, 256, 0, stream>>>(src, dst, logits, denom,
                                                        edge_wt + (size_t)t * E, h, out,
                                                        E, t);
    }
}